// GPUTransforms_78709570666904
// MI455X (gfx1250) — compile-verified
//
#include <hip/hip_runtime.h>
#include <stdint.h>

// ---------------- problem constants (from reference) ----------------
#define BATCH 64
#define CHN   3
#define HH    384
#define WW    384
#define PLANE (HH*WW)            // 147456
#define IMG   (CHN*PLANE)        // 442368 floats per image
#define CHUNK 8192               // floats per TDM tile (32 KB); 8192 | PLANE
#define CHUNKS_PER_IMG (IMG/CHUNK)        // 54
#define GRIDX 9
#define CHUNKS_PER_BLOCK (CHUNKS_PER_IMG/GRIDX)  // 6
#define V4_PER_CHUNK (CHUNK/4)   // 2048
#define NTHREADS 256

#define P_ERASE 0.5f
#define SCALE_LO 0.02f
#define SCALE_HI 0.33f
#define RATIO_LO 0.3f
#define RATIO_HI 3.3f

typedef __attribute__((ext_vector_type(4))) float        fx4;       // native vec4
typedef __attribute__((ext_vector_type(4))) unsigned int uint32x4_t;
typedef __attribute__((ext_vector_type(8))) int          int32x8_t;
typedef __attribute__((ext_vector_type(4))) int          int32x4_t;

// ---------------- counter-based RNG (deterministic) ----------------
__device__ __forceinline__ unsigned mix64(unsigned long long x) {
    x ^= x >> 33; x *= 0xff51afd7ed558ccdULL;
    x ^= x >> 33; x *= 0xc4ceb9fe1a85ec53ULL;
    x ^= x >> 33;
    return (unsigned)x;
}

__device__ __forceinline__ float u01(unsigned long long k) {
    // uniform in (0,1): never exactly 0 (safe for log)
    return (float)(mix64(k) >> 8) * 0x1.0p-24f + 0x1.0p-25f;
}

// standard normal via Box-Muller on a unique 64-bit element counter
__device__ __forceinline__ float gauss(unsigned long long e) {
    float u1 = u01(e * 2ull + 0x9E3779B97F4A7C15ULL);
    float u2 = u01(e * 2ull + 1ull + 0xD1B54A32D192ED03ULL);
    float r  = __builtin_sqrtf(-2.0f * __logf(u1));
    return r * __cosf(6.28318530717958647f * u2);
}

// ---------------- kernel 1: per-sample erase parameters ----------------
__global__ void erase_params_kernel(int* __restrict__ params) {
    int b = threadIdx.x;
    if (b >= BATCH) return;
    const unsigned long long seed = 0x2A5F0B4E14C3ull;  // fixed "key 42" analogue
    float up = u01(seed ^ (0x1ull << 40) ^ (unsigned long long)b);
    float ua = u01(seed ^ (0x2ull << 40) ^ (unsigned long long)b);
    float ur = u01(seed ^ (0x3ull << 40) ^ (unsigned long long)b);
    float ut = u01(seed ^ (0x4ull << 40) ^ (unsigned long long)b);
    float ul = u01(seed ^ (0x5ull << 40) ^ (unsigned long long)b);

    bool  apply = up <= P_ERASE;
    float area  = (float)PLANE;
    float ta    = (SCALE_LO + ua * (SCALE_HI - SCALE_LO)) * area;
    float ar    = RATIO_LO + ur * (RATIO_HI - RATIO_LO);
    int   he    = (int)roundf(sqrtf(ta * ar));
    int   we    = (int)roundf(sqrtf(ta / ar));
    int   valid = (apply && he < HH && we < WW) ? 1 : 0;
    int top = 0, left = 0;
    if (valid) {
        top  = (int)floorf(ut * (float)(HH - he + 1));
        left = (int)floorf(ul * (float)(WW - we + 1));
    } else {
        he = 0; we = 0;
    }
    int* p = params + b * 8;
    p[0] = valid; p[1] = top; p[2] = top + he; p[3] = left; p[4] = left + we;
    p[5] = 0; p[6] = 0; p[7] = 0;
}

// ---------------- TDM: DMA one contiguous 32KB tile HBM -> LDS ----------------
__device__ __forceinline__ void tdm_load_chunk(const void* gptr, unsigned lds_byte) {
    unsigned long long ga = (unsigned long long)(uintptr_t)gptr;
    uint32x4_t g0;
    g0[0] = 1u;                                              // count=1, user descriptor
    g0[1] = lds_byte;                                        // LDS byte address
    g0[2] = (unsigned)(ga & 0xffffffffull);                  // global addr lo
    g0[3] = (unsigned)((ga >> 32) & 0x1ffffffull) | (2u << 30); // addr hi | type=2
    int32x8_t g1;
    g1[0] = (2 << 16);          // workgroup_mask=0 (no cluster), data_size=4B
    g1[1] = (CHUNK << 16);      // tensor_dim0 lo16 = 8192
    g1[2] = (1 << 16);          // tensor_dim0 hi16 = 0, tensor_dim1 lo16 = 1
    g1[3] = (CHUNK << 16);      // tensor_dim1 hi16 = 0, tile_dim0 = 8192
    g1[4] = 1;                  // tile_dim1 = 1, tile_dim2 = 0
    g1[5] = CHUNK;              // tensor_dim0_stride lo32
    g1[6] = 0;                  // stride hi16 / dim1_stride lo16
    g1[7] = 0;                  // tensor_dim1_stride hi32
    int32x4_t z4 = {0, 0, 0, 0};                // groups 2/3 unused (2D tile)
    int32x8_t z8 = {0, 0, 0, 0, 0, 0, 0, 0};    // extra group (clang-23 6-arg form)
    __builtin_amdgcn_tensor_load_to_lds(g0, g1, z4, z4, z8, 0);
}

// ---------------- kernel 2: normalize + erase, TDM double-buffered ----------------
__global__ __launch_bounds__(NTHREADS)
void normalize_erase_kernel(const float* __restrict__ x,
                            const float* __restrict__ mean,
                            const float* __restrict__ stdv,
                            const int*   __restrict__ params,
                            float*       __restrict__ out) {
    __shared__ float lbuf[2][CHUNK];                    // 64 KB double buffer
    const int b   = blockIdx.y;
    const int tid = threadIdx.x;
    const bool isWave0 = (tid < 32);                    // wave-uniform on wave32

    // block-uniform erase params
    const int* pp = params + b * 8;
    const int valid = pp[0], top = pp[1], bot = pp[2], left = pp[3], right = pp[4];

    // fused normalize: y = x*inv_std + bias
    float inv_std[3], bias[3];
#pragma unroll
    for (int c = 0; c < 3; ++c) {
        float s = stdv[c];
        inv_std[c] = 1.0f / s;
        bias[c]    = -mean[c] / s;
    }

    const int firstChunk = blockIdx.x * CHUNKS_PER_BLOCK;
    const unsigned long long imgBase = (unsigned long long)b * IMG;
    const float* xBase = x + imgBase + (unsigned long long)firstChunk * CHUNK;

    if (isWave0) tdm_load_chunk(xBase, (unsigned)(uintptr_t)&lbuf[0][0]);

#pragma unroll 1   // keep one copy of the pipeline body (no 6x code bloat / I$ blowout)
    for (int ci = 0; ci < CHUNKS_PER_BLOCK; ++ci) {
        if (isWave0) __builtin_amdgcn_s_wait_tensorcnt(0);   // chunk ci landed in LDS
        __syncthreads();                                     // publish to all waves
        if (isWave0 && (ci + 1 < CHUNKS_PER_BLOCK))          // prefetch next chunk:
            tdm_load_chunk(xBase + (unsigned long long)(ci + 1) * CHUNK,
                           (unsigned)(uintptr_t)&lbuf[(ci + 1) & 1][0]);
            // safe: everyone passed the barrier, so no reader of buf[(ci+1)&1] remains

        const float* buf = lbuf[ci & 1];
        const int cid = firstChunk + ci;            // chunk index within image
        const int c   = cid / (PLANE / CHUNK);      // channel (PLANE/CHUNK = 18)
        const int cip = cid % (PLANE / CHUNK);      // chunk index within plane
        const float sc = inv_std[c], bi = bias[c];
        float* outp = out + imgBase + (unsigned long long)cid * CHUNK;
        const unsigned long long ebase = imgBase + (unsigned long long)cid * CHUNK;

#pragma unroll
        for (int j = 0; j < V4_PER_CHUNK / NTHREADS; ++j) {  // 8 float4 per thread
            const int v4 = j * NTHREADS + tid;              // vec4 index in chunk
            const int e0 = v4 * 4;                          // element offset in chunk
            fx4 xv = ((const fx4*)buf)[v4];                 // ds_load_b128
            fx4 r;
            r.x = xv.x * sc + bi;
            r.y = xv.y * sc + bi;
            r.z = xv.z * sc + bi;
            r.w = xv.w * sc + bi;
            if (valid) {
                const int p    = cip * CHUNK + e0;  // plane-local index
                const int hrow = p / WW;            // vec4 never crosses a row (384%4==0)
                if (hrow >= top && hrow < bot) {
                    const int wcol = p - hrow * WW;
#pragma unroll
                    for (int k = 0; k < 4; ++k) {
                        const int wc = wcol + k;
                        if (wc >= left && wc < right)
                            r[k] = gauss(ebase + (unsigned long long)(e0 + k));
                    }
                }
            }
            // streaming store: written once, never re-read -> keep out of L2's way
            __builtin_nontemporal_store(r, ((fx4*)outp) + v4);  // global_store_b128 (NT)
        }
        __syncthreads();   // all reads of buf[ci&1] done before it is refilled in ci+2
    }
}

// ---------------- launch ----------------
extern "C" void kernel_launch(void* const* d_in, const int* in_sizes, int n_in,
                              void* d_out, int out_size, void* d_ws, size_t ws_size,
                              hipStream_t stream) {
    const float* x    = (const float*)d_in[0];
    const float* mean = (const float*)d_in[1];
    const float* stdv = (const float*)d_in[2];
    float* out  = (float*)d_out;
    int* params = (int*)d_ws;                      // 64*8 ints = 2 KB scratch

    erase_params_kernel<<<1, BATCH, 0, stream>>>(params);

    dim3 grid(GRIDX, BATCH);
    normalize_erase_kernel<<<grid, NTHREADS, 0, stream>>>(x, mean, stdv, params, out);
}